// RankOnePlanesNew_89240830476841
// MI455X (gfx1250) — compile-verified
//
#include <hip/hip_runtime.h>

typedef __attribute__((ext_vector_type(2))) float v2f;
typedef __attribute__((ext_vector_type(8))) float v8f;

#define RANGE_INV  2.5f    // 1/0.4
#define RANGE_INV2 6.25f   // 1/0.4^2

// ---------------------------------------------------------------------------
// Kernel A: zero the 3x128 occupancy array (workspace persists across replays)
// ---------------------------------------------------------------------------
__global__ __launch_bounds__(384) void rop_zero_occ(int* __restrict__ occ) {
    occ[threadIdx.x] = 0;
}

// ---------------------------------------------------------------------------
// Kernel B: streaming occupancy scan over all B*N coordinate triples.
// 50 MB coalesced read; scattered stores of constant 1 (race-free).
// jnp.round == round-half-to-even == rintf (default RNE mode).
// ---------------------------------------------------------------------------
__global__ __launch_bounds__(256) void rop_occ_scan(const float* __restrict__ coords,
                                                    int* __restrict__ occ, int total) {
    int j = blockIdx.x * blockDim.x + threadIdx.x;
    if (j >= total) return;
    float x = coords[3 * j + 0];
    float y = coords[3 * j + 1];
    float z = coords[3 * j + 2];
    int k = (int)rintf((x + 2.0f) * 32.0f);   // scale = Nl/D = 128/4
    int l = (int)rintf((y + 2.0f) * 32.0f);
    int m = (int)rintf((z + 2.0f) * 32.0f);
    if ((unsigned)k < 128u) occ[k]       = 1;
    if ((unsigned)l < 128u) occ[128 + l] = 1;
    if ((unsigned)m < 128u) occ[256 + m] = 1;
}

// ---------------------------------------------------------------------------
// Kernel C: one wave32 does the entire MLP with V_WMMA_F32_16X16X4_F32.
//   feats (128x32 f32) @ w1^T (32x128) -> H (128x128), +b1, ReLU,
//   then dot each row with w2 (128) and +b2 -> small[0..127].
//   small[128] = c0 = relu(b1) . w2 + b2  (value for all rows >= 128).
// A-frag: lane holds M=(lane&15); {a.x,a.y} = K chunk {0,1} (lanes 0-15)
//         or {2,3} (lanes 16-31). B-frag symmetric with lane = N.
// D-frag: c8[v] = D[M = v + 8*(lane>>4)][N = lane&15].
// ---------------------------------------------------------------------------
__global__ __launch_bounds__(32) void rop_mlp_wmma(
    const float* __restrict__ lines0, const float* __restrict__ lines1,
    const float* __restrict__ lines2, const float* __restrict__ w1,
    const float* __restrict__ b1, const float* __restrict__ w2,
    const float* __restrict__ b2, const int* __restrict__ occ,
    float* __restrict__ small) {
    __shared__ float feats[128][32];
    __shared__ float w1s[128][32];
    const int lane = threadIdx.x;

    // Build feats rows 0..127 (32 lanes = 32 channels per row) + stage w1.
    for (int r = 0; r < 128; ++r) {
        float ox = occ[r]       ? 1.0f : 0.0f;
        float oy = occ[128 + r] ? 1.0f : 0.0f;
        float oz = occ[256 + r] ? 1.0f : 0.0f;
        float xe = lines0[r * 32 + lane] * ox;
        float ye = lines1[r * 32 + lane] * oy;
        float ze = lines2[r * 32 + lane] * oz;
        feats[r][lane] = (xe + ye + ze)
                       + (xe * ye + xe * ze + ye * ze) * RANGE_INV
                       + (xe * ye * ze) * RANGE_INV2;
        w1s[r][lane] = w1[r * 32 + lane];
    }
    __syncthreads();

    const float bias2 = b2[0];
    const int hh    = lane >> 4;     // 0 or 1 (lane half)
    const int l15   = lane & 15;
    const int khalf = hh * 2;        // K offset within 4-chunk for this half

    for (int mt = 0; mt < 8; ++mt) {
        float acc[8];
        #pragma unroll
        for (int v = 0; v < 8; ++v) acc[v] = 0.0f;
        const int arow = mt * 16 + l15;

        for (int nt = 0; nt < 8; ++nt) {
            const int brow = nt * 16 + l15;
            v8f c8 = {};
            #pragma unroll
            for (int kt = 0; kt < 8; ++kt) {           // K = 32 in steps of 4
                const int k0 = kt * 4 + khalf;
                v2f a; a.x = feats[arow][k0]; a.y = feats[arow][k0 + 1];
                v2f b; b.x = w1s[brow][k0];   b.y = w1s[brow][k0 + 1];
                c8 = __builtin_amdgcn_wmma_f32_16x16x4_f32(
                        false, a, false, b, (short)0, c8, false, false);
            }
            // Fused bias + ReLU + contraction with w2 along the hidden dim.
            const float bn = b1[nt * 16 + l15];
            const float wn = w2[nt * 16 + l15];
            #pragma unroll
            for (int v = 0; v < 8; ++v) {
                float h = c8[v] + bn;
                h = h > 0.0f ? h : 0.0f;
                acc[v] += h * wn;
            }
        }
        // Reduce over the 16 lanes of each half (each half owns 8 M-rows).
        #pragma unroll
        for (int v = 0; v < 8; ++v) {
            float s = acc[v];
            s += __shfl_xor(s, 1, 32);
            s += __shfl_xor(s, 2, 32);
            s += __shfl_xor(s, 4, 32);
            s += __shfl_xor(s, 8, 32);
            if (l15 == 0) small[mt * 16 + hh * 8 + v] = s + bias2;
        }
    }

    // Constant output for all rows >= 128 (feats row == 0 -> h = relu(b1)).
    float p = 0.0f;
    for (int h = lane; h < 128; h += 32) {
        float t = b1[h];
        t = t > 0.0f ? t : 0.0f;
        p += t * w2[h];
    }
    p += __shfl_xor(p, 1, 32);
    p += __shfl_xor(p, 2, 32);
    p += __shfl_xor(p, 4, 32);
    p += __shfl_xor(p, 8, 32);
    p += __shfl_xor(p, 16, 32);
    if (lane == 0) small[128] = p + bias2;
}

// ---------------------------------------------------------------------------
// Kernel D: broadcast results to the full (N,1) output. float4 stores.
// ---------------------------------------------------------------------------
__global__ __launch_bounds__(256) void rop_fill(const float* __restrict__ small,
                                                float* __restrict__ out, int n) {
    int t = blockIdx.x * blockDim.x + threadIdx.x;
    int i = t * 4;
    if (i >= n) return;
    const float c0 = small[128];
    if (i + 3 < n) {
        float4 v;
        if (i < 128) v = make_float4(small[i], small[i + 1], small[i + 2], small[i + 3]);
        else         v = make_float4(c0, c0, c0, c0);
        *(float4*)(out + i) = v;
    } else {
        for (int q = i; q < n; ++q) out[q] = (q < 128) ? small[q] : c0;
    }
}

// ---------------------------------------------------------------------------
extern "C" void kernel_launch(void* const* d_in, const int* in_sizes, int n_in,
                              void* d_out, int out_size, void* d_ws, size_t ws_size,
                              hipStream_t stream) {
    const float* coords = (const float*)d_in[0];  // (B, N, 3)
    const float* lines0 = (const float*)d_in[1];  // (128, 32)
    const float* lines1 = (const float*)d_in[2];
    const float* lines2 = (const float*)d_in[3];
    const float* w1     = (const float*)d_in[4];  // (128, 32)
    const float* b1     = (const float*)d_in[5];  // (128,)
    const float* w2     = (const float*)d_in[6];  // (1, 128)
    const float* b2     = (const float*)d_in[7];  // (1,)
    float*       out    = (float*)d_out;          // (N, 1) f32

    const int total = in_sizes[0] / 3;            // B*N triples (4M)
    int*   occ   = (int*)d_ws;                    // 3*128 ints
    float* small = (float*)((char*)d_ws + 2048);  // 129 floats

    rop_zero_occ<<<1, 384, 0, stream>>>(occ);
    rop_occ_scan<<<(total + 255) / 256, 256, 0, stream>>>(coords, occ, total);
    rop_mlp_wmma<<<1, 32, 0, stream>>>(lines0, lines1, lines2, w1, b1, w2, b2, occ, small);
    const int n4 = (out_size + 3) / 4;
    rop_fill<<<(n4 + 255) / 256, 256, 0, stream>>>(small, out, out_size);
}